// AugmentedGeometryScaledDotProductAttention_76974403878930
// MI455X (gfx1250) — compile-verified
//
#include <hip/hip_runtime.h>
#include <hip/hip_bf16.h>

typedef __attribute__((ext_vector_type(2))) float v2f;
typedef __attribute__((ext_vector_type(8))) float v8f;
typedef __attribute__((ext_vector_type(4))) unsigned int v4u;
typedef __attribute__((ext_vector_type(8))) int v8i;
typedef __attribute__((ext_vector_type(4))) int v4i;

#define BB 16
#define NN 256
#define HH 8
#define DM 512

// LDS row stride for the TDM-staged X tile: 512 dwords + 4-dword pad every 256
#define XROW 520

// ---------------------------------------------------------------------------
// Kernel 1: geometric embedding + per-head relu(fc_g) -> log(clip(g,1e-6))
// One thread per (b,i,j) pair. Writes staged log-g into the mn region of d_out
// (attn_fused reads it before overwriting the same addresses with softmax).
// ---------------------------------------------------------------------------
__global__ void geo_g_kernel(const float* __restrict__ boxes,
                             const float* __restrict__ Wg,
                             const float* __restrict__ bg,
                             float* __restrict__ lg_out) {
  __shared__ float sWg[HH * 64];
  __shared__ float sBg[HH];
  for (int t = threadIdx.x; t < HH * 64; t += blockDim.x) sWg[t] = Wg[t];
  if (threadIdx.x < HH) sBg[threadIdx.x] = bg[threadIdx.x];
  __syncthreads();

  int idx = blockIdx.x * blockDim.x + threadIdx.x;   // < B*N*N
  int j = idx & (NN - 1);
  int i = (idx >> 8) & (NN - 1);
  int b = idx >> 16;

  const float* bi = boxes + ((size_t)b * NN + i) * 4;
  const float* bj = boxes + ((size_t)b * NN + j) * 4;
  float cxi = (bi[0] + bi[2]) * 0.5f, cyi = (bi[1] + bi[3]) * 0.5f;
  float wi = bi[2] - bi[0] + 1.0f,    hi = bi[3] - bi[1] + 1.0f;
  float cxj = (bj[0] + bj[2]) * 0.5f, cyj = (bj[1] + bj[3]) * 0.5f;
  float wj = bj[2] - bj[0] + 1.0f,    hj = bj[3] - bj[1] + 1.0f;

  float pos[4];
  pos[0] = __logf(fmaxf(fabsf((cxi - cxj) / wi), 1e-3f));
  pos[1] = __logf(fmaxf(fabsf((cyi - cyj) / hi), 1e-3f));
  pos[2] = __logf(wi / wj);
  pos[3] = __logf(hi / hj);

  float sum[HH];
#pragma unroll
  for (int h = 0; h < HH; ++h) sum[h] = 0.0f;

#pragma unroll
  for (int f = 0; f < 8; ++f) {
    float dm = __powf(1000.0f, -(float)f * 0.125f);   // 1/WAVE_LEN^(f/8)
#pragma unroll
    for (int kp = 0; kp < 4; ++kp) {
      float m = 100.0f * pos[kp] * dm;
      float s, c;
      __sincosf(m, &s, &c);
      int gs = kp * 8 + f;          // sin block (first 32 features)
      int gc = 32 + kp * 8 + f;     // cos block
#pragma unroll
      for (int h = 0; h < HH; ++h)
        sum[h] += s * sWg[h * 64 + gs] + c * sWg[h * 64 + gc];
    }
  }

#pragma unroll
  for (int h = 0; h < HH; ++h) {
    float g = fmaxf(sum[h] + sBg[h], 0.0f);           // relu
    float l = __logf(fmaxf(g, 1e-6f));                // log(clip)
    lg_out[(((size_t)b * HH + h) * NN + i) * NN + j] = l;
  }
}

// ---------------------------------------------------------------------------
// Kernel 2: Y[4096,512] = X[4096,512] @ W[512,512]^T + bias, fp32 WMMA.
// The 16x512 X strip shared by all 8 waves of the block is staged in LDS ONCE
// via the Tensor Data Mover (async DMA, TENSORcnt), with hardware pad
// insertion (4 dwords per 256) so rows land at 520-dword stride (bank-spread,
// 8B aligned). W rows (distinct per wave) stream from global/L2.
// A 16x4 tile:  lane L holds (M=L%16, K=2*(L/16)+{0,1})  -> contiguous float2
// B 4x16 tile:  lane L holds (K=2*(L/16)+{0,1}, N=L%16)  -> contiguous float2
// D 16x16:      vgpr r, lane L -> (M = r + 8*(L/16), N = L%16)
// ---------------------------------------------------------------------------
__global__ void gemm4096x512x512(const float* __restrict__ X,
                                 const float* __restrict__ W,
                                 const float* __restrict__ bias,
                                 float* __restrict__ Y) {
  const int K = 512, Nd = 512;
  __shared__ float sX[16 * XROW + 8];   // +8 slack for trailing pad

  int lane = threadIdx.x & 31;
  int wv   = threadIdx.x >> 5;
  int lm   = lane & 15;
  int half = lane >> 4;

  int nTile = blockIdx.x * 8 + wv;    // 0..31
  int mTile = blockIdx.y;             // 0..255

  if (wv == 0) {
    // ---- Tensor DMA descriptor (D#): 16 rows x 512 f32, row-major, padded ----
    unsigned long long gaddr =
        (unsigned long long)(uintptr_t)(X + (size_t)mTile * 16 * K);
    unsigned ldsaddr = (unsigned)(uintptr_t)&sX[0];   // low 32 bits = LDS offset

    v4u g0;
    g0[0] = 1u;                                   // count=1 (valid), user mode
    g0[1] = ldsaddr;                              // lds_addr (bytes)
    g0[2] = (unsigned)(gaddr & 0xFFFFFFFFu);      // global_addr[31:0]
    g0[3] = (unsigned)((gaddr >> 32) & 0x01FFFFFFu) | (2u << 30);  // [56:32] | type=2

    v8i g1;
    // data_size=2 (4B), pad_enable, pad_interval=7 (256 dw), pad_amount=3 (4 dw)
    g1[0] = (int)((2u << 16) | (1u << 20) | (7u << 22) | (3u << 25));
    g1[1] = (int)((512u & 0xFFFFu) << 16);        // tensor_dim0[15:0]=512 @bits[63:48]
    g1[2] = (int)((4096u & 0xFFFFu) << 16);       // dim0 hi=0 | tensor_dim1 lo=4096
    g1[3] = (int)(512u << 16);                    // dim1 hi=0 | tile_dim0=512
    g1[4] = 16;                                   // tile_dim1=16, tile_dim2=0
    g1[5] = 512;                                  // tensor_dim0_stride lo32
    g1[6] = 0;                                    // stride hi | dim1_stride lo
    g1[7] = 0;

    v4i g2 = {0, 0, 0, 0};                        // unused (2D tensor)
    v4i g3 = {0, 0, 0, 0};
    v8i g4 = {0, 0, 0, 0, 0, 0, 0, 0};            // unused trailing group

    __builtin_amdgcn_tensor_load_to_lds(g0, g1, g2, g3, g4, 0);
    __builtin_amdgcn_s_wait_tensorcnt((short)0);
  }
  __syncthreads();

  const float* wrow = W + (size_t)(nTile * 16 + lm) * K + 2 * half;
  const float* xl   = &sX[lm * XROW + 2 * half];

  v8f acc = {};
#pragma unroll 8
  for (int kk = 0; kk < 256; kk += 4) {
    v2f a  = *(const v2f*)(xl + kk);
    v2f bb = *(const v2f*)(wrow + kk);
    acc = __builtin_amdgcn_wmma_f32_16x16x4_f32(false, a, false, bb,
                                                (short)0, acc, false, false);
  }
#pragma unroll 8
  for (int kk = 256; kk < 512; kk += 4) {
    v2f a  = *(const v2f*)(xl + kk + 4);          // skip 4-dword TDM pad @256
    v2f bb = *(const v2f*)(wrow + kk);
    acc = __builtin_amdgcn_wmma_f32_16x16x4_f32(false, a, false, bb,
                                                (short)0, acc, false, false);
  }

  int n = nTile * 16 + lm;
  float bn = bias[n];
#pragma unroll
  for (int r = 0; r < 8; ++r) {
    int m = mTile * 16 + r + 8 * half;
    Y[(size_t)m * Nd + n] = acc[r] + bn;
  }
}

// ---------------------------------------------------------------------------
// Kernel 3: fused attention for one (b, h, 16-query-row block).
// Phase 1: S = QK^T/8 + log_g (WMMA over d=64), into LDS [16][272].
// Phase 2: row softmax in LDS, write mn to d_out.
// Phase 3: P @ V (WMMA over k=256, A tiles read from LDS) -> attn_out ws.
// ---------------------------------------------------------------------------
__global__ void attn_fused(const float* __restrict__ Q,
                           const float* __restrict__ Km,
                           const float* __restrict__ V,
                           float* __restrict__ mn,     // [B,H,N,N]; log_g on entry
                           float* __restrict__ AO) {   // [B*N, 512]
  __shared__ float sS[16][272];     // 272 pad keeps rows 8B-aligned
  __shared__ float sred[16][16];
  __shared__ float srow[16];

  int tid  = threadIdx.x;
  int lane = tid & 31, wv = tid >> 5;
  int lm   = lane & 15, half = lane >> 4;
  int b = blockIdx.z, h = blockIdx.y, qt = blockIdx.x;

  const size_t hoff   = (size_t)h * 64;
  const float* qbase  = Q + ((size_t)(b * NN + qt * 16 + lm)) * DM + hoff + 2 * half;
  const size_t mnBase = (((size_t)b * HH + h) * NN + qt * 16) * NN;

  // ---- phase 1: scores ----
  for (int jt = wv; jt < 16; jt += 8) {
    const float* kbase = Km + ((size_t)(b * NN + jt * 16 + lm)) * DM + hoff + 2 * half;
    v8f acc = {};
#pragma unroll
    for (int kk = 0; kk < 64; kk += 4) {
      v2f a  = *(const v2f*)(qbase + kk);
      v2f bb = *(const v2f*)(kbase + kk);
      acc = __builtin_amdgcn_wmma_f32_16x16x4_f32(false, a, false, bb,
                                                  (short)0, acc, false, false);
    }
#pragma unroll
    for (int r = 0; r < 8; ++r) {
      int i = r + 8 * half;
      int j = jt * 16 + lm;
      float lg = mn[mnBase + (size_t)i * NN + j];
      sS[i][j] = acc[r] * 0.125f + lg;      // 1/sqrt(64)
    }
  }
  __syncthreads();

  // ---- phase 2: softmax over 256 keys; 16 threads cooperate per row ----
  {
    int row = tid >> 4;
    int seg = tid & 15;
    float mx = -3.4e38f;
#pragma unroll
    for (int t = 0; t < 16; ++t) mx = fmaxf(mx, sS[row][seg * 16 + t]);
    sred[row][seg] = mx;
    __syncthreads();
    if (seg == 0) {
      float m2 = sred[row][0];
      for (int t = 1; t < 16; ++t) m2 = fmaxf(m2, sred[row][t]);
      srow[row] = m2;
    }
    __syncthreads();
    float rm = srow[row];
    float sum = 0.0f;
#pragma unroll
    for (int t = 0; t < 16; ++t) {
      float e = __expf(sS[row][seg * 16 + t] - rm);
      sS[row][seg * 16 + t] = e;
      sum += e;
    }
    sred[row][seg] = sum;
    __syncthreads();
    if (seg == 0) {
      float s2 = 0.0f;
      for (int t = 0; t < 16; ++t) s2 += sred[row][t];
      srow[row] = 1.0f / s2;
    }
    __syncthreads();
    float inv = srow[row];
#pragma unroll
    for (int t = 0; t < 16; ++t) {
      int j = seg * 16 + t;
      float p = sS[row][j] * inv;
      sS[row][j] = p;
      mn[mnBase + (size_t)row * NN + j] = p;  // overwrite staged log_g with mn
    }
  }
  __syncthreads();

  // ---- phase 3: P @ V ; waves 0..3 each own a 16-wide d tile ----
  if (wv < 4) {
    int dt = wv;
    const float* vbase = V + ((size_t)b * NN) * DM + hoff + dt * 16 + lm;
    v8f acc = {};
    for (int kk = 0; kk < NN; kk += 4) {
      int k0 = kk + 2 * half;
      v2f a = *(const v2f*)&sS[lm][k0];
      v2f bb;
      bb.x = vbase[(size_t)k0 * DM];
      bb.y = vbase[(size_t)(k0 + 1) * DM];
      acc = __builtin_amdgcn_wmma_f32_16x16x4_f32(false, a, false, bb,
                                                  (short)0, acc, false, false);
    }
#pragma unroll
    for (int r = 0; r < 8; ++r) {
      int i = qt * 16 + r + 8 * half;
      AO[((size_t)b * NN + i) * DM + hoff + dt * 16 + lm] = acc[r];
    }
  }
}

// ---------------------------------------------------------------------------
extern "C" void kernel_launch(void* const* d_in, const int* in_sizes, int n_in,
                              void* d_out, int out_size, void* d_ws, size_t ws_size,
                              hipStream_t stream) {
  (void)in_sizes; (void)n_in; (void)out_size; (void)ws_size;

  const float* queries = (const float*)d_in[0];
  const float* keys    = (const float*)d_in[1];
  const float* values  = (const float*)d_in[2];
  const float* boxes   = (const float*)d_in[3];
  const float* Wq      = (const float*)d_in[4];
  const float* bq      = (const float*)d_in[5];
  const float* Wk      = (const float*)d_in[6];
  const float* bk      = (const float*)d_in[7];
  const float* Wv      = (const float*)d_in[8];
  const float* bv      = (const float*)d_in[9];
  const float* Wo      = (const float*)d_in[10];
  const float* bo      = (const float*)d_in[11];
  const float* Wg      = (const float*)d_in[12];
  const float* bg      = (const float*)d_in[13];

  float* out = (float*)d_out;                              // [B,N,512]
  float* mn  = (float*)d_out + (size_t)BB * NN * DM;       // [B,H,N,N]

  float* ws = (float*)d_ws;
  const size_t MAT = (size_t)BB * NN * DM;                 // 4096*512
  float* q  = ws;
  float* k  = ws + MAT;
  float* v  = ws + 2 * MAT;
  float* ao = ws + 3 * MAT;

  // 1) staged log-g into mn region
  geo_g_kernel<<<dim3(BB * NN * NN / 256), 256, 0, stream>>>(boxes, Wg, bg, mn);

  // 2) Q/K/V projections (fp32 WMMA GEMM, TDM-staged A tiles)
  dim3 gg(4, 256), gb(256);
  gemm4096x512x512<<<gg, gb, 0, stream>>>(queries, Wq, bq, q);
  gemm4096x512x512<<<gg, gb, 0, stream>>>(keys,    Wk, bk, k);
  gemm4096x512x512<<<gg, gb, 0, stream>>>(values,  Wv, bv, v);

  // 3) fused attention: scores + softmax (-> mn) + P@V (-> ao)
  attn_fused<<<dim3(NN / 16, HH, BB), 256, 0, stream>>>(q, k, v, mn, ao);

  // 4) output projection
  gemm4096x512x512<<<gg, gb, 0, stream>>>(ao, Wo, bo, out);
}